// GraphAttentionLayer_10608569221524
// MI455X (gfx1250) — compile-verified
//
#include <hip/hip_runtime.h>

#define NN      8192
#define FIN     128
#define FOUT    64
#define ALPHA   0.2f
#define NEG_BIG (-9.0e15f)
#define NWAVES  4
#define JCHUNK  (NN / NWAVES)   // columns per wave

typedef float v2f __attribute__((ext_vector_type(2)));
typedef float v8f __attribute__((ext_vector_type(8)));

// ---------------------------------------------------------------------------
// Kernel 1: wh = h_nodes(8192x128) @ weight(128x64).  134 MFLOP -> negligible.
// One block per node row, weight stays hot in L2/WGP$.
// ---------------------------------------------------------------------------
__global__ __launch_bounds__(64)
void wh_kernel(const float* __restrict__ h, const float* __restrict__ w,
               float* __restrict__ wh)
{
    const int i = blockIdx.x;
    const int j = threadIdx.x;
    __shared__ float hrow[FIN];
    hrow[j]      = h[(long)i * FIN + j];
    hrow[j + 64] = h[(long)i * FIN + 64 + j];
    __syncthreads();
    float s = 0.0f;
    #pragma unroll 8
    for (int f = 0; f < FIN; ++f)
        s = fmaf(hrow[f], w[f * FOUT + j], s);
    wh[(long)i * FOUT + j] = s;
}

// ---------------------------------------------------------------------------
// Kernel 2: src[i] = wh[i,:].a1 ; dst[i] = wh[i,:].a2
// ---------------------------------------------------------------------------
__global__ __launch_bounds__(256)
void srcdst_kernel(const float* __restrict__ wh, const float* __restrict__ att,
                   float* __restrict__ srcv, float* __restrict__ dstv)
{
    const int i = blockIdx.x * blockDim.x + threadIdx.x;
    const float* wr = wh + (long)i * FOUT;
    float s1 = 0.0f, s2 = 0.0f;
    #pragma unroll 8
    for (int f = 0; f < FOUT; ++f) {
        const float v = wr[f];
        s1 = fmaf(v, att[f], s1);
        s2 = fmaf(v, att[FOUT + f], s2);
    }
    srcv[i] = s1;
    dstv[i] = s2;
}

// ---------------------------------------------------------------------------
// Kernel 3: fused mask + leaky_relu + online softmax + (attention @ wh) + ELU.
// Block = 16 output rows, 4 waves; wave w streams columns [w*2048,(w+1)*2048).
// adj is read exactly once (268 MB total -> HBM roofline floor).
//
// WMMA f32 16x16x4 A-layout: VGPR0 = (M=lane, K=0 | K=2), VGPR1 = (K=1 | K=3).
// So lane l (<16) computes row l columns with j%4 in {0,1}; lane l+16 computes
// row l columns with j%4 in {2,3} -> P feeds the A operand straight from regs.
// C-layout: element r -> row r (lanes 0-15) / row 8+r (lanes 16-31).
// ---------------------------------------------------------------------------
__global__ __launch_bounds__(32 * NWAVES)
void gat_fused_kernel(const int* __restrict__ adj,
                      const float* __restrict__ wh,
                      const float* __restrict__ srcv,
                      const float* __restrict__ dstv,
                      float* __restrict__ out)
{
    __shared__ float s_acc[NWAVES][32][33];   // +1 pad vs 64-bank LDS
    __shared__ float s_m[NWAVES][16];
    __shared__ float s_l[NWAVES][16];

    const int lane = threadIdx.x & 31;
    const int wv   = threadIdx.x >> 5;
    const int lr   = lane & 15;               // row within 16-row tile
    const int hk   = lane >> 4;               // 0: K%4 in {0,1}; 1: {2,3}
    const int r0   = blockIdx.x * 16;
    const int row  = r0 + lr;
    const int jbeg = wv * JCHUNK;
    const int jend = jbeg + JCHUNK;

    const float s_i  = srcv[row];
    const long  arow = (long)row * NN;

    float m    = -3.0e38f;                    // running row max
    float lsum = 0.0f;                        // per-lane partial row sum
    v8f acc0 = {0.f,0.f,0.f,0.f,0.f,0.f,0.f,0.f};
    v8f acc1 = {0.f,0.f,0.f,0.f,0.f,0.f,0.f,0.f};
    v8f acc2 = {0.f,0.f,0.f,0.f,0.f,0.f,0.f,0.f};
    v8f acc3 = {0.f,0.f,0.f,0.f,0.f,0.f,0.f,0.f};

    for (int j0 = jbeg; j0 < jend; j0 += 64) {
        if (j0 + 1024 < jend)
            __builtin_prefetch(adj + arow + j0 + 1024, 0, 0);

        // ---- masked leaky_relu scores for this lane's 32 columns ----------
        float e[32];
        float chmax = -3.0e38f;
        #pragma unroll
        for (int t = 0; t < 16; ++t) {
            const int j = j0 + 4 * t + 2 * hk;
            const int2   a2i = *(const int2*)  (adj + arow + j);
            const float2 d2  = *(const float2*)(dstv + j);
            float e0 = s_i + d2.x; e0 = (e0 > 0.f) ? e0 : ALPHA * e0;
            float e1 = s_i + d2.y; e1 = (e1 > 0.f) ? e1 : ALPHA * e1;
            e0 = (a2i.x > 0) ? e0 : NEG_BIG;
            e1 = (a2i.y > 0) ? e1 : NEG_BIG;
            e[2 * t]     = e0;
            e[2 * t + 1] = e1;
            chmax = fmaxf(chmax, fmaxf(e0, e1));
        }

        // ---- online softmax: row max is shared by the lane pair (l, l+16) -
        chmax = fmaxf(chmax, __shfl_xor(chmax, 16, 32));
        const float mnew  = fmaxf(m, chmax);
        const float scale = __expf(m - mnew);
        m = mnew;

        // rescale accumulators: C element r belongs to row r / 8+r
        #pragma unroll
        for (int r = 0; r < 8; ++r) {
            const float sr = __shfl(scale, hk ? (8 + r) : r, 32);
            acc0[r] *= sr; acc1[r] *= sr; acc2[r] *= sr; acc3[r] *= sr;
        }
        lsum *= scale;

        #pragma unroll
        for (int t = 0; t < 32; ++t) {
            const float p = __expf(e[t] - m);
            e[t] = p;
            lsum += p;
        }

        // ---- C(16x64) += P(16x64) @ Wh(64x64) via f32 WMMA, 16 K-steps ----
        #pragma unroll
        for (int t = 0; t < 16; ++t) {
            const int k = j0 + 4 * t + 2 * hk;   // this lane's K rows of B
            v2f a; a[0] = e[2 * t]; a[1] = e[2 * t + 1];
            const float* w0 = wh + (long)k * FOUT;
            const float* w1 = w0 + FOUT;
            v2f b0, b1, b2, b3;
            b0[0] = w0[lr];      b0[1] = w1[lr];
            b1[0] = w0[16 + lr]; b1[1] = w1[16 + lr];
            b2[0] = w0[32 + lr]; b2[1] = w1[32 + lr];
            b3[0] = w0[48 + lr]; b3[1] = w1[48 + lr];
            acc0 = __builtin_amdgcn_wmma_f32_16x16x4_f32(false, a, false, b0, (short)0, acc0, false, false);
            acc1 = __builtin_amdgcn_wmma_f32_16x16x4_f32(false, a, false, b1, (short)0, acc1, false, false);
            acc2 = __builtin_amdgcn_wmma_f32_16x16x4_f32(false, a, false, b2, (short)0, acc2, false, false);
            acc3 = __builtin_amdgcn_wmma_f32_16x16x4_f32(false, a, false, b3, (short)0, acc3, false, false);
        }
    }

    // ---- publish per-wave partials --------------------------------------
    lsum += __shfl_xor(lsum, 16, 32);         // full row sum for this wave
    #pragma unroll
    for (int r = 0; r < 8; ++r) {
        s_acc[wv][lane][r]      = acc0[r];
        s_acc[wv][lane][8 + r]  = acc1[r];
        s_acc[wv][lane][16 + r] = acc2[r];
        s_acc[wv][lane][24 + r] = acc3[r];
    }
    if (lane < 16) { s_m[wv][lr] = m; s_l[wv][lr] = lsum; }
    __syncthreads();

    if (wv != 0) return;

    // ---- combine NWAVES partial softmaxes, normalize, ELU, store --------
    #pragma unroll
    for (int r = 0; r < 8; ++r) {
        const int rr = r + 8 * hk;            // row slot of acc element r
        float mt = s_m[0][rr];
        #pragma unroll
        for (int w = 1; w < NWAVES; ++w) mt = fmaxf(mt, s_m[w][rr]);
        float ltot = 0.f, a0 = 0.f, a1 = 0.f, a2 = 0.f, a3 = 0.f;
        #pragma unroll
        for (int w = 0; w < NWAVES; ++w) {
            const float sc = __expf(s_m[w][rr] - mt);
            ltot += s_l[w][rr] * sc;
            a0 += s_acc[w][lane][r]      * sc;
            a1 += s_acc[w][lane][8 + r]  * sc;
            a2 += s_acc[w][lane][16 + r] * sc;
            a3 += s_acc[w][lane][24 + r] * sc;
        }
        const float li = 1.0f / fmaxf(ltot, 1e-30f);
        float o0 = a0 * li, o1 = a1 * li, o2 = a2 * li, o3 = a3 * li;
        o0 = (o0 > 0.f) ? o0 : __expf(o0) - 1.f;   // ELU (alpha=1)
        o1 = (o1 > 0.f) ? o1 : __expf(o1) - 1.f;
        o2 = (o2 > 0.f) ? o2 : __expf(o2) - 1.f;
        o3 = (o3 > 0.f) ? o3 : __expf(o3) - 1.f;
        const long ob = (long)(r0 + rr) * FOUT + lr;
        out[ob]      = o0;
        out[ob + 16] = o1;
        out[ob + 32] = o2;
        out[ob + 48] = o3;
    }
}

// ---------------------------------------------------------------------------
extern "C" void kernel_launch(void* const* d_in, const int* in_sizes, int n_in,
                              void* d_out, int out_size, void* d_ws, size_t ws_size,
                              hipStream_t stream)
{
    const float* h   = (const float*)d_in[0];
    // d_in[1] = edge: unused by the reference computation
    const int*   adj = (const int*)  d_in[2];
    const float* w   = (const float*)d_in[3];
    const float* att = (const float*)d_in[4];
    float* out = (float*)d_out;

    float* wh   = (float*)d_ws;                  // 8192*64 f32  (2 MB)
    float* srcv = wh + (size_t)NN * FOUT;        // 8192 f32
    float* dstv = srcv + NN;                     // 8192 f32

    wh_kernel     <<<NN,        64,          0, stream>>>(h, w, wh);
    srcdst_kernel <<<NN / 256,  256,         0, stream>>>(wh, att, srcv, dstv);
    gat_fused_kernel<<<NN / 16, 32 * NWAVES, 0, stream>>>(adj, wh, srcv, dstv, out);
}